// PrivateNLMs_61048665145444
// MI455X (gfx1250) — compile-verified
//
#include <hip/hip_runtime.h>

typedef float v2f __attribute__((ext_vector_type(2)));
typedef float v8f __attribute__((ext_vector_type(8)));

#define B_SZ 1024
#define D_SZ 2048
#define M_SZ 64
#define H_SZ 128
#define HP   144  // float2 row stride: (2*HP)%64 == 32 -> wave halves hit disjoint bank sets

// Async global->LDS staging path (gfx1250 GLOBAL_LOAD_ASYNC_TO_LDS_B32, ASYNCcnt).
#if defined(__has_builtin)
#if __has_builtin(__builtin_amdgcn_global_load_async_to_lds_b32) && \
    __has_builtin(__builtin_amdgcn_s_wait_asynccnt)
#define HAVE_ASYNC_LDS 1
#endif
#endif

typedef __attribute__((address_space(1))) int as1_int;  // global ("__device__") int*
typedef __attribute__((address_space(3))) int as3_int;  // LDS ("__shared__") int*

// gelu(x) = 0.5*x*(1 + tanh(0.79788456*(x + 0.044715*x^3)))  computed as
// x - x*rcp(E+1),  E = exp2(x*(0.10294514*x^2 + 2.30214049));  5 VALU + 2 TRANS.
__device__ __forceinline__ float fast_gelu(float x) {
    float t = x * x;
    float u = __builtin_fmaf(t, 0.10294514f, 2.30214049f);
    float s = x * u;
    float E = __builtin_amdgcn_exp2f(s);
    float r = __builtin_amdgcn_rcpf(E + 1.0f);
    return __builtin_fmaf(-x, r, x);
}

__global__ __launch_bounds__(256) void fused_nlm_kernel(
    const float* __restrict__ a_hist, const float* __restrict__ w1,
    const float* __restrict__ b1, const float* __restrict__ w2,
    const float* __restrict__ b2, float* __restrict__ z)
{
    // w1[:, :, d] stored as K-pairs: w1p[m/2][h] = (w1[2m'][h], w1[2m'+1][h])
    // so a WMMA B fragment is ONE ds_load_b64 into an even-aligned VGPR pair.
    __shared__ v2f   w1p[(M_SZ / 2) * HP];
    __shared__ float w2s[H_SZ];
    __shared__ float b1s[H_SZ];

    const int d   = blockIdx.x;
    const int tid = threadIdx.x;

    // ---- Stage w1[:, :, d] (64x128), w2[:, d], b1[:, d] into LDS ----
    // 4B-strided gathers; w1 (64MB) is L2-resident (192MB L2) so these hit L2.
#if HAVE_ASYNC_LDS
    for (int i = tid; i < (M_SZ / 2) * H_SZ; i += 256) {
        int m2 = i >> 7;
        int h  = i & (H_SZ - 1);
        const float* g0 = w1 + (size_t)((2 * m2) * H_SZ + h) * D_SZ + d;
        const float* g1 = w1 + (size_t)((2 * m2 + 1) * H_SZ + h) * D_SZ + d;
        float* l = (float*)&w1p[m2 * HP + h];
        __builtin_amdgcn_global_load_async_to_lds_b32((as1_int*)g0, (as3_int*)(l + 0), 0, 0);
        __builtin_amdgcn_global_load_async_to_lds_b32((as1_int*)g1, (as3_int*)(l + 1), 0, 0);
    }
    if (tid < H_SZ) {
        const float* gw2 = w2 + (size_t)tid * D_SZ + d;
        const float* gb1 = b1 + (size_t)tid * D_SZ + d;
        __builtin_amdgcn_global_load_async_to_lds_b32((as1_int*)gw2, (as3_int*)&w2s[tid], 0, 0);
        __builtin_amdgcn_global_load_async_to_lds_b32((as1_int*)gb1, (as3_int*)&b1s[tid], 0, 0);
    }
    __builtin_amdgcn_s_wait_asynccnt(0);
    __syncthreads();
#else
    for (int i = tid; i < (M_SZ / 2) * H_SZ; i += 256) {
        int m2 = i >> 7;
        int h  = i & (H_SZ - 1);
        v2f p;
        p.x = w1[(size_t)((2 * m2) * H_SZ + h) * D_SZ + d];
        p.y = w1[(size_t)((2 * m2 + 1) * H_SZ + h) * D_SZ + d];
        w1p[m2 * HP + h] = p;
    }
    if (tid < H_SZ) {
        w2s[tid] = w2[(size_t)tid * D_SZ + d];
        b1s[tid] = b1[(size_t)tid * D_SZ + d];
    }
    __syncthreads();
#endif

    const int lane   = tid & 31;
    const int wid    = tid >> 5;
    const int half   = lane >> 4;   // 0: lanes 0-15, 1: lanes 16-31
    const int ln     = lane & 15;
    const int b_base = blockIdx.y * 256 + wid * 32;  // 2 b-tiles of 16 per wave

    // Preload A fragments for both 16x64 tiles.
    // WMMA A 16x4 f32 layout: lane -> (M = lane&15, K = 2*(lane>>4) + {0,1})
    v2f afrag[2][16];
#pragma unroll
    for (int bt = 0; bt < 2; ++bt) {
        const size_t row = (size_t)(b_base + bt * 16 + ln) * D_SZ + d;
        const float* ap = a_hist + row * M_SZ + half * 2;
#pragma unroll
        for (int kk = 0; kk < 16; ++kk)
            afrag[bt][kk] = *(const v2f*)(ap + kk * 4);
    }

    float part[2][8];
#pragma unroll
    for (int bt = 0; bt < 2; ++bt)
#pragma unroll
        for (int r = 0; r < 8; ++r) part[bt][r] = 0.0f;

#pragma unroll
    for (int t = 0; t < 8; ++t) {          // 8 h-tiles of 16
        const int col = t * 16 + ln;
        // WMMA B 4x16 f32 layout: lane -> (K = 2*(lane>>4) + {0,1}, N = lane&15)
        // = exactly the float2 at w1p[2*kk + half][col]: one ds_load_b64, no repacking.
        v2f bfrag[16];
#pragma unroll
        for (int kk = 0; kk < 16; ++kk)
            bfrag[kk] = w1p[(kk * 2 + half) * HP + col];

        const float w2v  = w2s[col];
        const float bias = b1s[col];

#pragma unroll
        for (int bt = 0; bt < 2; ++bt) {
            v8f acc = {0.f, 0.f, 0.f, 0.f, 0.f, 0.f, 0.f, 0.f};
#pragma unroll
            for (int kk = 0; kk < 16; ++kk) {
                acc = __builtin_amdgcn_wmma_f32_16x16x4_f32(
                    false, afrag[bt][kk], false, bfrag[kk],
                    (short)0, acc, false, false);
            }
            // C layout: VGPR r, lane -> (M = r + 8*(lane>>4), N = lane&15) = (b row, h col)
#pragma unroll
            for (int r = 0; r < 8; ++r) {
                float g = fast_gelu(acc[r] + bias);
                part[bt][r] = __builtin_fmaf(g, w2v, part[bt][r]);
            }
        }
    }

    // Reduce over h%16 (the N dim): sum across each 16-lane half.
#pragma unroll
    for (int bt = 0; bt < 2; ++bt)
#pragma unroll
        for (int r = 0; r < 8; ++r) {
#pragma unroll
            for (int off = 1; off < 16; off <<= 1)
                part[bt][r] += __shfl_xor(part[bt][r], off, 32);
        }

    if (ln == 0) {  // lanes 0 and 16 each write 16 z scalars
        const float bz = b2[d];
#pragma unroll
        for (int bt = 0; bt < 2; ++bt)
#pragma unroll
            for (int r = 0; r < 8; ++r) {
                int b = b_base + bt * 16 + half * 8 + r;
                z[(size_t)b * D_SZ + d] = part[bt][r] + bz;
            }
    }
}

extern "C" void kernel_launch(void* const* d_in, const int* in_sizes, int n_in,
                              void* d_out, int out_size, void* d_ws, size_t ws_size,
                              hipStream_t stream) {
    const float* a_hist = (const float*)d_in[0];
    const float* w1     = (const float*)d_in[1];
    const float* b1     = (const float*)d_in[2];
    const float* w2     = (const float*)d_in[3];
    const float* b2     = (const float*)d_in[4];
    float* zout = (float*)d_out;

    dim3 grid(D_SZ, B_SZ / 256);  // 2048 x 4 workgroups
    fused_nlm_kernel<<<grid, dim3(256), 0, stream>>>(a_hist, w1, b1, w2, b2, zout);
}